// TransD_61933428417129
// MI455X (gfx1250) — compile-verified
//
#include <hip/hip_runtime.h>

// TransD scoring, algebraically reduced:
//   out[i,:] = rp_i * (dot(hp_i,h_i) - dot(tp_i,t_i)) + h_i - t_i + r_i
// Memory-bound gather kernel; the per-row dot products are computed exactly in
// f32 on the matrix pipe via chained V_WMMA_F32_16X16X4_F32 (diagonal of
// Hp*H^T - Tp*T^T for a 16-row tile), everything else is coalesced float4
// streaming.

typedef __attribute__((ext_vector_type(2))) float v2f;
typedef __attribute__((ext_vector_type(8))) float v8f;

#define DIM 128
#define WAVES_PER_BLOCK 4
#define BLOCK_THREADS (WAVES_PER_BLOCK * 32)
#define ROWS_PER_WAVE 16
#define ROWS_PER_BLOCK (WAVES_PER_BLOCK * ROWS_PER_WAVE)

__global__ __launch_bounds__(BLOCK_THREADS) void transd_wmma_kernel(
    const int* __restrict__ head,
    const int* __restrict__ relation,
    const int* __restrict__ tail,
    const float* __restrict__ ent_emb,
    const float* __restrict__ ent_map_emb,
    const float* __restrict__ rel_emb,
    const float* __restrict__ rel_map_emb,
    float* __restrict__ out,
    int batch)
{
    __shared__ float sdot[WAVES_PER_BLOCK][ROWS_PER_WAVE];

    const int lane = threadIdx.x & 31;
    const int wv   = threadIdx.x >> 5;
    const int tileBase = blockIdx.x * ROWS_PER_BLOCK + wv * ROWS_PER_WAVE;

    const int m    = lane & 15;   // tile row handled by this lane (both halves)
    const int half = lane >> 4;   // 0 or 1 (selects K sub-columns in WMMA layout)

    int row = tileBase + m;
    if (row >= batch) row = batch - 1;  // clamp reads; final stores are guarded

    const int hIdx32 = head[row];
    const int tIdx32 = tail[row];
    const int rIdx32 = relation[row];

    const float* __restrict__ hRow  = ent_emb     + (long)hIdx32 * DIM;
    const float* __restrict__ hpRow = ent_map_emb + (long)hIdx32 * DIM;
    const float* __restrict__ tRow  = ent_emb     + (long)tIdx32 * DIM;
    const float* __restrict__ tpRow = ent_map_emb + (long)tIdx32 * DIM;

    // Warm the (L2-resident) relation rows needed in the output phase.
    __builtin_prefetch(rel_emb     + (long)rIdx32 * DIM, 0, 3);
    __builtin_prefetch(rel_map_emb + (long)rIdx32 * DIM, 0, 3);

    // ---- Phase 1: diag(Hp*H^T) - diag(Tp*T^T) via chained f32 WMMA ----
    // A 16x4 f32 layout: lane L holds A[L%16, kk + 2*(L>>4) + {0,1}] in v[0],v[1]
    // B  4x16 f32 layout: lane L holds B[2*(L>>4) + {0,1}, L%16]     in v[0],v[1]
    // With B = H^T both operands use identical per-lane addressing.
    v8f acc = {0.f, 0.f, 0.f, 0.f, 0.f, 0.f, 0.f, 0.f};
    const int cb = half * 2;

#pragma unroll
    for (int kk = 0; kk < DIM; kk += 4) {
        v2f ah = *(const v2f*)(hpRow + kk + cb);
        v2f bh = *(const v2f*)(hRow  + kk + cb);
        v2f at = *(const v2f*)(tpRow + kk + cb);
        v2f bt = *(const v2f*)(tRow  + kk + cb);
        at = -at;  // f32 WMMA has no A/B neg modifier; fold the subtraction here
        acc = __builtin_amdgcn_wmma_f32_16x16x4_f32(
            false, ah, false, bh, (short)0, acc, false, false);
        acc = __builtin_amdgcn_wmma_f32_16x16x4_f32(
            false, at, false, bt, (short)0, acc, false, false);
    }

    // ---- Phase 2: extract the 16 diagonal values and broadcast via LDS ----
    // C/D 16x16 f32 layout: VGPR v, lane L -> row M = v + 8*(L>>4), col N = L%16.
    // Element (n,n): lane (n<8 ? n : n+16), VGPR n&7.
    {
        float dval = acc[0];
#pragma unroll
        for (int q = 1; q < 8; ++q)
            if ((m & 7) == q) dval = acc[q];
        if (half == (m >> 3))
            sdot[wv][m] = dval;
    }
    __syncthreads();

    // ---- Phase 3: streamed output, fully coalesced (512B/row/matrix) ----
#pragma unroll 4
    for (int mm = 0; mm < ROWS_PER_WAVE; ++mm) {
        const int gRow = tileBase + mm;
        const int hI = __shfl(hIdx32, mm, 32);
        const int tI = __shfl(tIdx32, mm, 32);
        const int rI = __shfl(rIdx32, mm, 32);
        const float d = sdot[wv][mm];

        const float4 hv = ((const float4*)(ent_emb     + (long)hI * DIM))[lane];
        const float4 tv = ((const float4*)(ent_emb     + (long)tI * DIM))[lane];
        const float4 rv = ((const float4*)(rel_emb     + (long)rI * DIM))[lane];
        const float4 pv = ((const float4*)(rel_map_emb + (long)rI * DIM))[lane];

        float4 o;
        o.x = fmaf(pv.x, d, hv.x - tv.x + rv.x);
        o.y = fmaf(pv.y, d, hv.y - tv.y + rv.y);
        o.z = fmaf(pv.z, d, hv.z - tv.z + rv.z);
        o.w = fmaf(pv.w, d, hv.w - tv.w + rv.w);

        if (gRow < batch)
            ((float4*)(out + (long)gRow * DIM))[lane] = o;
    }
}

extern "C" void kernel_launch(void* const* d_in, const int* in_sizes, int n_in,
                              void* d_out, int out_size, void* d_ws, size_t ws_size,
                              hipStream_t stream) {
    const int*   head        = (const int*)d_in[0];
    const int*   relation    = (const int*)d_in[1];
    const int*   tail        = (const int*)d_in[2];
    const float* ent_emb     = (const float*)d_in[3];
    const float* ent_map_emb = (const float*)d_in[4];
    const float* rel_emb     = (const float*)d_in[5];
    const float* rel_map_emb = (const float*)d_in[6];
    float* out = (float*)d_out;

    const int batch = in_sizes[0];  // 16384
    const int grid  = (batch + ROWS_PER_BLOCK - 1) / ROWS_PER_BLOCK;

    transd_wmma_kernel<<<dim3(grid), dim3(BLOCK_THREADS), 0, stream>>>(
        head, relation, tail, ent_emb, ent_map_emb, rel_emb, rel_map_emb,
        out, batch);
}